// NFM_25855703122475
// MI455X (gfx1250) — compile-verified
//
#include <hip/hip_runtime.h>
#include <hip/hip_bf16.h>
#include <stdint.h>

// FM layer: out[b] = 0.5*((sum_j v*e)^2 - sum_j (v*e)^2) . h + sum_j w[idx]*v + b0
// B=8192, NNZ=64, D=64. Memory-bound sparse gather -> async global->LDS pipeline.
//
// Per wave (1 wave = 1 batch row):
//  - 32x global_load_async_to_lds_b128 issued up front (2 embed rows / op,
//    16KB in flight per wave, ASYNCcnt <= 32 < 63 max). LDS staging is wave-
//    private: no workgroup barriers, only s_wait_asynccnt gates (in-order).
//  - lanes 0-15 consume even rows, lanes 16-31 odd rows, 4 dims per lane via
//    one ds_load_b128 per chunk; halves merged with shfl_xor(16) before the
//    square (s and q are linear sums, so the merge is exact).
//  - idx/val broadcasts use v_readlane (VALU) instead of ds_bpermute to keep
//    the LDS pipe free for the staged embedding reads.

#define B_ROWS 8192
#define NNZ 64
#define WAVES_PER_BLOCK 4
#define THREADS (WAVES_PER_BLOCK * 32)
#define CHUNKS 32                       // 2 embedding rows per chunk
#define WAVE_LDS_FLOATS (CHUNKS * 128)  // 32 slots * 512B = 16KB per wave

__global__ __launch_bounds__(THREADS) void fm_gather_kernel(
    const float* __restrict__ vals,    // [B, NNZ]
    const float* __restrict__ embed,   // [NFEAT, D]
    const float* __restrict__ wlin,    // [NFEAT]
    const float* __restrict__ blin,    // [1]
    const float* __restrict__ hwt,     // [D] (ones, but honor it)
    const int*   __restrict__ idxs,    // [B, NNZ]
    float*       __restrict__ out)     // [B]
{
    __shared__ __align__(16) float smem[WAVES_PER_BLOCK * WAVE_LDS_FLOATS]; // 64 KB

    const int lane = threadIdx.x & 31;
    const int wave = threadIdx.x >> 5;
    const int row  = blockIdx.x * WAVES_PER_BLOCK + wave;

    const int base = row * NNZ;
    // Each lane owns nnz slots (lane) and (lane+32). Coalesced loads.
    const int   idx_a = idxs[base + lane];
    const int   idx_b = idxs[base + 32 + lane];
    const float val_a = vals[base + lane];
    const float val_b = vals[base + 32 + lane];

    // Linear term partial: scattered 4B loads, w_linear (2MB) is L2-resident.
    float lin = fmaf(wlin[idx_a], val_a, wlin[idx_b] * val_b);

    // Workgroup-relative LDS byte offset of this wave's staging region.
    const unsigned ldsWaveBase =
        (unsigned)(uintptr_t)(&smem[wave * WAVE_LDS_FLOATS]);

    const int      half     = lane >> 4;                  // 0: even row, 1: odd row
    const unsigned laneByte = (unsigned)(lane & 15) << 4; // 16B per lane

    // ---- Issue ALL 32 async gather ops up front. SADDR form: 32-bit voffset
    //      (table is 128MB < 2^31), scalar base in SGPR pair.
#pragma unroll
    for (int c = 0; c < CHUNKS; ++c) {
        int i0 = __builtin_amdgcn_readlane((c < 16) ? idx_a : idx_b, (2 * c) & 31);
        int i1 = __builtin_amdgcn_readlane((c < 16) ? idx_a : idx_b, (2 * c + 1) & 31);
        unsigned sel  = (unsigned)(half ? i1 : i0);
        unsigned voff = (sel << 8) + laneByte;            // idx*256B + lane slot
        unsigned lo   = ldsWaveBase + ((unsigned)c << 9) + ((unsigned)half << 8) + laneByte;
        asm volatile("global_load_async_to_lds_b128 %0, %1, %2"
                     :: "v"(lo), "v"(voff), "s"(embed) : "memory");
    }

    // ---- Consume: lane owns dims 4*(lane&15) .. +3 of its half's rows.
    float s0 = 0.f, s1 = 0.f, s2 = 0.f, s3 = 0.f;
    float q0 = 0.f, q1 = 0.f, q2 = 0.f, q3 = 0.f;
    const char* wbase = (const char*)&smem[wave * WAVE_LDS_FLOATS] +
                        ((unsigned)half << 8) + laneByte;

#pragma unroll
    for (int g = 0; g < 4; ++g) {
        // Async loads complete in order: gate each group of 8 chunks.
        if      (g == 0) asm volatile("s_wait_asynccnt 24" ::: "memory");
        else if (g == 1) asm volatile("s_wait_asynccnt 16" ::: "memory");
        else if (g == 2) asm volatile("s_wait_asynccnt 8"  ::: "memory");
        else             asm volatile("s_wait_asynccnt 0"  ::: "memory");
#pragma unroll
        for (int cc = 0; cc < 8; ++cc) {
            const int c = g * 8 + cc;
            // Broadcast v for even/odd row of this chunk via readlane (VALU).
            float ve = __int_as_float(__builtin_amdgcn_readlane(
                __float_as_int((c < 16) ? val_a : val_b), (2 * c) & 31));
            float vo = __int_as_float(__builtin_amdgcn_readlane(
                __float_as_int((c < 16) ? val_a : val_b), (2 * c + 1) & 31));
            float vj = half ? vo : ve;

            float4 e = *(const float4*)(wbase + ((unsigned)c << 9)); // ds_load_b128
            s0 = fmaf(vj, e.x, s0);  s1 = fmaf(vj, e.y, s1);
            s2 = fmaf(vj, e.z, s2);  s3 = fmaf(vj, e.w, s3);
            float w0 = vj * e.x, w1 = vj * e.y, w2 = vj * e.z, w3 = vj * e.w;
            q0 = fmaf(w0, w0, q0);  q1 = fmaf(w1, w1, q1);
            q2 = fmaf(w2, w2, q2);  q3 = fmaf(w3, w3, q3);
        }
    }

    // Merge even/odd halves (s, q are linear sums -> exact before squaring).
    s0 += __shfl_xor(s0, 16, 32);  s1 += __shfl_xor(s1, 16, 32);
    s2 += __shfl_xor(s2, 16, 32);  s3 += __shfl_xor(s3, 16, 32);
    q0 += __shfl_xor(q0, 16, 32);  q1 += __shfl_xor(q1, 16, 32);
    q2 += __shfl_xor(q2, 16, 32);  q3 += __shfl_xor(q3, 16, 32);

    // z = 0.5*(s^2 - q); y = z . h. Each dim now lives in two lanes (lane,
    // lane^16), so the 32-lane reduce double-counts FM: use 0.25 instead of 0.5.
    const float4 hv = *((const float4*)hwt + (lane & 15));
    float fm = fmaf(s0, s0, -q0) * hv.x + fmaf(s1, s1, -q1) * hv.y +
               fmaf(s2, s2, -q2) * hv.z + fmaf(s3, s3, -q3) * hv.w;
    float part = 0.25f * fm + lin;   // lin is distinct per lane (counted once)

    // wave32 all-reduce
#pragma unroll
    for (int off = 16; off; off >>= 1) part += __shfl_xor(part, off, 32);

    if (lane == 0) out[row] = part + blin[0];
}

extern "C" void kernel_launch(void* const* d_in, const int* in_sizes, int n_in,
                              void* d_out, int out_size, void* d_ws, size_t ws_size,
                              hipStream_t stream) {
    // setup_inputs() order:
    // 0: feature_vals [B,NNZ] f32      1: feature_embed [NFEAT,D] f32
    // 2: w_linear [NFEAT] f32          3: b_linear [1] f32
    // 4: h_weight [D] f32              5: feature_indices [B,NNZ] i32
    const float* vals  = (const float*)d_in[0];
    const float* embed = (const float*)d_in[1];
    const float* wlin  = (const float*)d_in[2];
    const float* blin  = (const float*)d_in[3];
    const float* hwt   = (const float*)d_in[4];
    const int*   idxs  = (const int*)d_in[5];
    float* out = (float*)d_out;

    dim3 grid(B_ROWS / WAVES_PER_BLOCK);   // 2048 blocks x 4 waves = 8192 rows
    dim3 block(THREADS);
    fm_gather_kernel<<<grid, block, 0, stream>>>(vals, embed, wlin, blin, hwt, idxs, out);
}